// SSM_model_24292335026628
// MI455X (gfx1250) — compile-verified
//
#include <hip/hip_runtime.h>
#include <math.h>

// ---------------------------------------------------------------------------
// S4/DPLR SSM forward on MI455X (gfx1250).
// FFT-convolution path (memory-bound, ~450MB traffic @ 23.3TB/s ≈ 20us)
// instead of 550-TFLOP time-domain matmul.
// - One workgroup per (batch,channel) sequence: full 16384-pt complex-f32 FFT
//   lives in 128KB of the 320KB WGP LDS (single pass, no global round trips).
// - First 4 FFT stages = permuted 16-pt DFT matrix applied as a 16x16 complex
//   matmul via V_WMMA_F32_16X16X4_F32 (f32 accuracy required over 14 stages).
// - All FFT loop bounds are compile-time: scalar (SALU) loop control, WMMA
//   always issues with EXEC all-ones as the ISA requires.
// - Strided (stride 256B) sequence gather done with the Tensor Data Mover
//   (tensor_load_to_lds, 6-arg form on this toolchain) in the transpose kernel.
// ---------------------------------------------------------------------------

#define LSEQ   16384      // sequence length L
#define NSTATE 64
#define NCH    64
#define NBATCH 16
#define NFFT   16384      // packed-complex length representing the 2L=32768 real conv
#define NF2    8192
#define KF_STRIDE 16392   // per-channel Kf stride (>= 16385), float2 elements
#define NTH    256        // 8 wave32 per workgroup

typedef float v2f __attribute__((ext_vector_type(2)));
typedef float v8f __attribute__((ext_vector_type(8)));
typedef unsigned int u32x4 __attribute__((ext_vector_type(4)));
typedef int i32x4 __attribute__((ext_vector_type(4)));
typedef int i32x8 __attribute__((ext_vector_type(8)));

// ---------------- complex helpers ----------------
__device__ __forceinline__ float2 cmul(float2 a, float2 b) {
  return make_float2(a.x * b.x - a.y * b.y, a.x * b.y + a.y * b.x);
}
__device__ __forceinline__ float2 cadd(float2 a, float2 b) { return make_float2(a.x + b.x, a.y + b.y); }
__device__ __forceinline__ float2 csub(float2 a, float2 b) { return make_float2(a.x - b.x, a.y - b.y); }
__device__ __forceinline__ float2 cconj(float2 a) { return make_float2(a.x, -a.y); }
__device__ __forceinline__ float2 cscale(float2 a, float s) { return make_float2(a.x * s, a.y * s); }
__device__ __forceinline__ float2 cdiv(float2 a, float2 b) {
  float d = b.x * b.x + b.y * b.y + 1e-30f;
  return make_float2((a.x * b.x + a.y * b.y) / d, (a.y * b.x - a.x * b.y) / d);
}
__device__ __forceinline__ float2 cexpi(float ang) {
  float s, c; __sincosf(ang, &s, &c); return make_float2(c, s);
}
__device__ __forceinline__ int bitrev4(int k) {
  return ((k & 1) << 3) | ((k & 2) << 1) | ((k & 4) >> 1) | ((k & 8) >> 3);
}

// ---------------- in-LDS iterative complex FFT -----------------------------
// buf: N complex values in LDS, exactly NTH(=256) threads.
// dirsign = -1 forward, +1 inverse (caller applies 1/N scaling).
// Stage 1: global bit-reversal. Stage 2 (WMMA): combined first 4 radix-2
// stages == multiply each group of 16 by the column-permuted 16-pt DFT matrix
// F[m][k] = exp(dirsign*2*pi*i*m*brev4(k)/16); batched 16 groups per WMMA tile
// (complex matmul = 4 real 16x16x16 matmuls = 16 chained 16x16x4 f32 WMMAs).
// Stage 3: remaining radix-2 stages (len = 32..N) in VALU.
// N and all trip counts are compile-time -> uniform scalar loop control.
template <int N, int LOGN>
__device__ void fft_lds(float2* buf, float dirsign) {
  const int tid = threadIdx.x;

#pragma unroll 1
  for (int it = 0; it < N / NTH; ++it) {
    int i = tid + it * NTH;
    unsigned j = __brev((unsigned)i) >> (32 - LOGN);
    if (j > (unsigned)i) { float2 t = buf[i]; buf[i] = buf[j]; buf[j] = t; }
  }
  __syncthreads();

#if defined(__gfx1250__) && __has_builtin(__builtin_amdgcn_wmma_f32_16x16x4_f32)
  {
    constexpr int NWAVES = NTH / 32;
    constexpr int NTILES = N >> 8;          // 16 groups of 16 elements per tile
    constexpr int PERWAVE = NTILES / NWAVES;
    const int lane = tid & 31;
    const int wave = tid >> 5;
    const int hi = lane >> 4;    // lane-half select (ISA 16x4 / 4x16 layouts)
    const int m  = lane & 15;    // output row (A) / column (B,D)
    // A fragments (DFT matrix), identical for every tile: precompute 4 k-steps.
    v2f a_re[4], a_im[4], a_imn[4];
#pragma unroll
    for (int ks = 0; ks < 4; ++ks) {
      int c0 = ks * 4 + (hi << 1);
      float2 w0 = cexpi(dirsign * 6.2831853f * (float)(m * bitrev4(c0))     * (1.0f / 16.0f));
      float2 w1 = cexpi(dirsign * 6.2831853f * (float)(m * bitrev4(c0 + 1)) * (1.0f / 16.0f));
      a_re[ks].x  =  w0.x; a_re[ks].y  =  w1.x;
      a_im[ks].x  =  w0.y; a_im[ks].y  =  w1.y;
      a_imn[ks].x = -w0.y; a_imn[ks].y = -w1.y;  // F32 WMMA has no A-neg modifier
    }
#pragma unroll 1
    for (int it = 0; it < PERWAVE; ++it) {   // constant trip count: EXEC stays full
      const int g0 = (wave + it * NWAVES) << 4;
      v8f cr, ci;
#pragma unroll
      for (int q = 0; q < 8; ++q) { cr[q] = 0.0f; ci[q] = 0.0f; }
#pragma unroll
      for (int ks = 0; ks < 4; ++ks) {
        // B fragment: X[k][n] = group (g0+n) element k, per ISA 4x16 layout.
        int r0 = ks * 4 + (hi << 1);
        float2 e0 = buf[(g0 + m) * 16 + r0];
        float2 e1 = buf[(g0 + m) * 16 + r0 + 1];
        v2f b_re, b_im;
        b_re.x = e0.x; b_re.y = e1.x;
        b_im.x = e0.y; b_im.y = e1.y;
        // Yr = Fr*Xr - Fi*Xi ; Yi = Fr*Xi + Fi*Xr
        cr = __builtin_amdgcn_wmma_f32_16x16x4_f32(false, a_re[ks],  false, b_re, (short)0, cr, false, false);
        cr = __builtin_amdgcn_wmma_f32_16x16x4_f32(false, a_imn[ks], false, b_im, (short)0, cr, false, false);
        ci = __builtin_amdgcn_wmma_f32_16x16x4_f32(false, a_re[ks],  false, b_im, (short)0, ci, false, false);
        ci = __builtin_amdgcn_wmma_f32_16x16x4_f32(false, a_im[ks],  false, b_re, (short)0, ci, false, false);
      }
      // D layout: VGPR v -> row v + 8*hi, column m. LDS ops are in-order per
      // wave, and tiles are wave-disjoint, so no extra barrier needed here.
#pragma unroll
      for (int v = 0; v < 8; ++v) {
        buf[(g0 + m) * 16 + (v + (hi << 3))] = make_float2(cr[v], ci[v]);
      }
    }
  }
  constexpr int LEN0 = 32;
#else
  constexpr int LEN0 = 2;
#endif
  __syncthreads();

#pragma unroll 1
  for (int len = LEN0; len <= N; len <<= 1) {
    const int half = len >> 1;
#pragma unroll 1
    for (int it = 0; it < N / (2 * NTH); ++it) {
      const int p = tid + it * NTH;
      const int blk = p / half;
      const int pos = p - blk * half;
      const int i0 = blk * len + pos;
      const int i1 = i0 + half;
      float2 tw = cexpi(dirsign * 6.2831853f * (float)pos / (float)len);
      float2 a  = buf[i0];
      float2 bb = cmul(tw, buf[i1]);
      buf[i0] = cadd(a, bb);
      buf[i1] = csub(a, bb);
    }
    __syncthreads();
  }
}

// real<->packed-complex spectrum helpers (rfft-of-2N via N-pt complex FFT)
__device__ __forceinline__ float2 unpack_r(float2 Za, float2 Zb, int j, int N) {
  // U[j] = A - i*W*B, W = e^{-i*pi*j/N}
  float2 A = cscale(cadd(Za, cconj(Zb)), 0.5f);
  float2 B = cscale(csub(Za, cconj(Zb)), 0.5f);
  float2 t = cmul(cexpi(-3.14159265f * (float)j / (float)N), B);
  return make_float2(A.x + t.y, A.y - t.x);
}
__device__ __forceinline__ float2 repack_r(float2 Ya, float2 Yb, int j, int N) {
  // Z[j] = A + i*V*B, V = e^{+i*pi*j/N}
  float2 A = cscale(cadd(Ya, cconj(Yb)), 0.5f);
  float2 B = cscale(csub(Ya, cconj(Yb)), 0.5f);
  float2 t = cmul(cexpi(3.14159265f * (float)j / (float)N), B);
  return make_float2(A.x - t.y, A.y + t.x);
}

// ---------------- 1) C_ = C @ (I - A_L), row 0 (rows identical) -------------
__global__ void crow_kernel(const float2* __restrict__ C, const float2* __restrict__ A_L,
                            float2* __restrict__ Crow) {
  int n = threadIdx.x;
  if (n >= NSTATE) return;
  float2 acc = make_float2(0.f, 0.f);
  for (int m = 0; m < NSTATE; ++m) {
    float2 e = make_float2(((m == n) ? 1.f : 0.f) - A_L[m * NSTATE + n].x,
                           -A_L[m * NSTATE + n].y);
    acc = cadd(acc, cmul(C[m], e));
  }
  Crow[n] = acc;
}

// ---------------- 2) Cauchy kernel K_w over 8193 frequencies ---------------
// (B,C,P,Q are tiled identically across channels -> one shared K_w row)
__global__ void kw_kernel(const float2* __restrict__ Crow, const float2* __restrict__ B,
                          const float2* __restrict__ P, const float2* __restrict__ Q,
                          const float2* __restrict__ eig, const float* __restrict__ stepp,
                          float2* __restrict__ Kw) {
  int k = blockIdx.x * blockDim.x + threadIdx.x;
  if (k > NF2) return;                                   // k = 0..8192
  float step = stepp[0];
  float2 z     = cexpi(-6.28318530718f * (float)k / (float)LSEQ);
  float2 onepz = make_float2(1.f + z.x, z.y);
  float2 onemz = make_float2(1.f - z.x, -z.y);
  float2 w     = cscale(cdiv(onemz, onepz), 2.f / step);
  float2 k00 = {0.f, 0.f}, k01 = {0.f, 0.f}, k10 = {0.f, 0.f}, k11 = {0.f, 0.f};
  for (int n = 0; n < NSTATE; ++n) {
    float2 d = csub(w, eig[n]);
    float den = d.x * d.x + d.y * d.y + 1e-30f;
    float2 r = make_float2(d.x / den, -d.y / den);       // resolvent 1/(w-eig)
    float2 Bn = B[n * NCH], Pn = P[n * NCH], Qn = Q[n * NCH];
    k00 = cadd(k00, cmul(cmul(Crow[n], Bn), r));
    k01 = cadd(k01, cmul(cmul(Crow[n], Pn), r));
    k10 = cadd(k10, cmul(cmul(cconj(Qn), Bn), r));
    k11 = cadd(k11, cmul(cmul(cconj(Qn), Pn), r));
  }
  float2 kw = csub(k00, cmul(cdiv(k01, make_float2(1.f + k11.x, k11.y)), k10));
  Kw[k] = cmul(cdiv(make_float2(2.f, 0.f), onepz), kw);
}

// ---------------- 3) Kf = rfft(pad(irfft(K_w), 2L)) per channel ------------
// Packed trick: irfft output pairs ARE the packed input of the forward pass.
__global__ void __launch_bounds__(NTH)
build_kf_kernel(const float2* __restrict__ Kw, float2* __restrict__ Kf) {
  extern __shared__ float2 buf[];                        // 16384 complex = 128 KB
  const int c = blockIdx.x;
  const int tid = threadIdx.x;
  // inverse packing: Z[k] from U=Kw[0..8192], then 8192-pt inverse FFT
  for (int k = tid; k < NF2; k += NTH)
    buf[k] = repack_r(Kw[k], Kw[NF2 - k], k, NF2);
  __syncthreads();
  fft_lds<NF2, 13>(buf, +1.0f);
  const float s = 1.0f / (float)NF2;
  for (int t = tid; t < NFFT; t += NTH)
    buf[t] = (t < NF2) ? cscale(buf[t], s) : make_float2(0.f, 0.f);
  __syncthreads();
  fft_lds<NFFT, 14>(buf, -1.0f);                         // forward 16384-pt
  float2* KfC = Kf + (size_t)c * KF_STRIDE;
  for (int k = tid; k <= NFFT / 2; k += NTH) {
    int mk = (NFFT - k) & (NFFT - 1);
    KfC[k] = unpack_r(buf[k], buf[mk], k, NFFT);
    if (k == 0)             KfC[NFFT]     = unpack_r(buf[0], buf[0], NFFT, NFFT);
    else if (k != NFFT / 2) KfC[NFFT - k] = unpack_r(buf[mk], buf[k], NFFT - k, NFFT);
  }
}

// ---------------- 4) transpose x (B,L,CH) -> xT (B,CH,L) via TDM -----------
__global__ void __launch_bounds__(NTH)
transpose_x_kernel(const float* __restrict__ x, float* __restrict__ xT) {
  __shared__ float tile[64 * 64];
  const int b = blockIdx.y;
  const int t0 = blockIdx.x * 64;
  const int tid = threadIdx.x;
#if defined(__gfx1250__) && __has_builtin(__builtin_amdgcn_tensor_load_to_lds) && __has_builtin(__builtin_amdgcn_s_wait_tensorcnt)
  if (tid == 0) {
    // D# per CDNA5 ISA: 2-D tensor, dim0=channel (contig),
    // dim1=time (stride CH), tile 64x64 elements, data_size=4B.
    unsigned long long ga = (unsigned long long)(const void*)(x + ((size_t)b * LSEQ + t0) * NCH);
    unsigned int ldsa = (unsigned int)(unsigned long long)(void*)&tile[0]; // low 32b = LDS offset
    u32x4 g0;
    g0[0] = 1u;                                                   // count=1
    g0[1] = ldsa;                                                 // lds_addr
    g0[2] = (unsigned int)(ga & 0xFFFFFFFFull);                   // global_addr lo
    g0[3] = (unsigned int)((ga >> 32) & 0x01FFFFFFull) | (2u << 30); // addr hi | type=2
    i32x8 g1;
    g1[0] = (2 << 16);                                            // data_size=4B
    g1[1] = (NCH & 0xFFFF) << 16;                                 // tensor_dim0 lo16
    g1[2] = ((NCH >> 16) & 0xFFFF) | ((LSEQ & 0xFFFF) << 16);     // dim0 hi | dim1 lo
    g1[3] = ((LSEQ >> 16) & 0xFFFF) | (64 << 16);                 // dim1 hi | tile_dim0
    g1[4] = 64;                                                   // tile_dim1 (tile_dim2=0)
    g1[5] = NCH;                                                  // tensor_dim0_stride lo32
    g1[6] = 0;
    g1[7] = 0;
    i32x4 z4; z4[0] = 0; z4[1] = 0; z4[2] = 0; z4[3] = 0;
    i32x8 z8; for (int q = 0; q < 8; ++q) z8[q] = 0;              // unused group (2D tensor)
    __builtin_amdgcn_tensor_load_to_lds(g0, g1, z4, z4, z8, 0);
    __builtin_amdgcn_s_wait_tensorcnt(0);
  }
  __syncthreads();
#else
  for (int i = 0; i < 16; ++i) {
    int r = (tid >> 6) + i * 4;
    int c = tid & 63;
    tile[r * 64 + c] = x[((size_t)b * LSEQ + t0 + r) * NCH + c];
  }
  __syncthreads();
#endif
  for (int i = 0; i < 16; ++i) {
    int c  = (tid >> 6) + i * 4;
    int tl = tid & 63;
    xT[((size_t)b * NCH + c) * LSEQ + t0 + tl] = tile[tl * 64 + c];
  }
}

// ---------------- 5) per-sequence FFT convolution (in place) ---------------
__global__ void __launch_bounds__(NTH)
conv_kernel(float* __restrict__ xT, const float2* __restrict__ Kf) {
  extern __shared__ float2 buf[];                        // 16384 complex = 128 KB
  const int c = blockIdx.x, b = blockIdx.y;
  const int tid = threadIdx.x;
  float* seq = xT + ((size_t)b * NCH + c) * LSEQ;
  // pack zero-padded real 2L signal into NFFT complex values
  for (int t = tid; t < NFFT; t += NTH)
    buf[t] = (t < NF2) ? make_float2(seq[2 * t], seq[2 * t + 1]) : make_float2(0.f, 0.f);
  __syncthreads();
  fft_lds<NFFT, 14>(buf, -1.0f);
  // spectral multiply: unpack real spectrum, multiply by Kf, repack.
  // Thread k owns index pair (k, N-k): in-place safe without extra barriers.
  const float2* KfC = Kf + (size_t)c * KF_STRIDE;
  for (int k = tid; k <= NFFT / 2; k += NTH) {
    int mk = (NFFT - k) & (NFFT - 1);
    float2 Zk = buf[k], Zm = buf[mk];
    float2 Yk = cmul(unpack_r(Zk, Zm, k, NFFT),        KfC[k]);
    float2 Ym = cmul(unpack_r(Zm, Zk, NFFT - k, NFFT), KfC[NFFT - k]);
    buf[k] = repack_r(Yk, Ym, k, NFFT);
    if (k != 0 && k != NFFT / 2) buf[mk] = repack_r(Ym, Yk, NFFT - k, NFFT);
  }
  __syncthreads();
  fft_lds<NFFT, 14>(buf, +1.0f);
  const float inv = 1.0f / (float)NFFT;
  for (int t = tid; t < NF2; t += NTH) {                 // first L outputs only
    float2 zt = buf[t];
    seq[2 * t]     = zt.x * inv;
    seq[2 * t + 1] = zt.y * inv;
  }
}

// ---------------- 6) transpose back + D*x + exact GELU ---------------------
__global__ void __launch_bounds__(NTH)
gelu_out_kernel(const float* __restrict__ yT, const float* __restrict__ x,
                const float* __restrict__ D, float* __restrict__ out) {
  __shared__ float tile[64 * 65];
  const int b = blockIdx.y, t0 = blockIdx.x * 64, tid = threadIdx.x;
  for (int i = 0; i < 16; ++i) {
    int c  = (tid >> 6) + i * 4;
    int tl = tid & 63;
    tile[c * 65 + tl] = yT[((size_t)b * NCH + c) * LSEQ + t0 + tl];
  }
  __syncthreads();
  const float d0 = D[0];
  for (int i = 0; i < 16; ++i) {
    int r = (tid >> 6) + i * 4;
    int c = tid & 63;
    size_t idx = ((size_t)b * LSEQ + t0 + r) * NCH + c;
    float v = tile[c * 65 + r] + d0 * x[idx];
    out[idx] = 0.5f * v * (1.0f + erff(v * 0.70710678118f));
  }
}

// ---------------------------------------------------------------------------
extern "C" void kernel_launch(void* const* d_in, const int* in_sizes, int n_in,
                              void* d_out, int out_size, void* d_ws, size_t ws_size,
                              hipStream_t stream) {
  const float*  x   = (const float*)d_in[0];
  const float2* A_L = (const float2*)d_in[1];
  const float2* B   = (const float2*)d_in[2];
  const float2* C   = (const float2*)d_in[3];
  const float2* P   = (const float2*)d_in[4];
  const float2* Q   = (const float2*)d_in[5];
  const float2* eig = (const float2*)d_in[6];
  const float*  stp = (const float*)d_in[7];
  const float*  D   = (const float*)d_in[8];
  float* out = (float*)d_out;

  char* w = (char*)d_ws;
  float* xT = (float*)w;                                           // 64 MB
  size_t off = (size_t)NBATCH * NCH * LSEQ * sizeof(float);
  float2* Kf = (float2*)(w + off); off += (size_t)NCH * KF_STRIDE * sizeof(float2);
  float2* Kw = (float2*)(w + off); off += 8200 * sizeof(float2);
  float2* Cr = (float2*)(w + off);

  const size_t fft_lds_bytes = (size_t)NFFT * sizeof(float2);      // 128 KB < 320 KB WGP LDS

  crow_kernel<<<1, 64, 0, stream>>>(C, A_L, Cr);
  kw_kernel<<<(NF2 + 1 + 255) / 256, 256, 0, stream>>>(Cr, B, P, Q, eig, stp, Kw);
  build_kf_kernel<<<NCH, NTH, fft_lds_bytes, stream>>>(Kw, Kf);
  transpose_x_kernel<<<dim3(LSEQ / 64, NBATCH), NTH, 0, stream>>>(x, xT);
  conv_kernel<<<dim3(NCH, NBATCH), NTH, fft_lds_bytes, stream>>>(xT, Kf);
  gelu_out_kernel<<<dim3(LSEQ / 64, NBATCH), NTH, 0, stream>>>(xT, x, D, out);
}